// DatasetTopK_24429773980153
// MI455X (gfx1250) — compile-verified
//
#include <hip/hip_runtime.h>

// ---------------------------------------------------------------------------
// DatasetTopK on MI455X (gfx1250, wave32).
//   Phase 1: fp32 WMMA (v_wmma_f32_16x16x4_f32) score tiles + fused-histogram
//            exact radix-select top-100 per (16-query tile, 16K-candidate chunk).
//            B-fragments preloaded to registers (one load clause per tile-pair),
//            split-K accumulators for 4-way WMMA ILP.
//   Phase 2: per-query merge of 64 partial top-100s + bitonic sort (descending).
// Workspace use: 1024 rows * 64 chunks * 100 keys * 4B = 26.2 MB.
// ---------------------------------------------------------------------------

#define QTILE   16
#define DIMS    64
#define SUBC    1024                  // candidates per sub-chunk (LDS resident)
#define NSUB    16
#define CHUNK   (SUBC * NSUB)         // 16384 candidates per block
#define NCHUNK  64                    // 1048576 / CHUNK
#define NQT     64                    // 1024 / QTILE
#define KTOP    100
#define POOLCAP 512
#define KPITCH  (SUBC + 4)            // pad to de-conflict LDS banks

typedef __attribute__((ext_vector_type(2))) float v2f;
typedef __attribute__((ext_vector_type(8))) float v8f;

__device__ __forceinline__ v8f wmma_f32_k4(v2f a, v2f b, v8f c) {
  // D = A(16x4 f32) * B(4x16 f32) + C(16x16 f32)
  return __builtin_amdgcn_wmma_f32_16x16x4_f32(false, a, false, b, (short)0, c,
                                               false, false);
}

// order-preserving float -> uint key (larger float => larger key)
__device__ __forceinline__ unsigned f2k(float x) {
  unsigned u = __float_as_uint(x);
  return (u & 0x80000000u) ? ~u : (u | 0x80000000u);
}
__device__ __forceinline__ float k2f(unsigned k) {
  return (k & 0x80000000u) ? __uint_as_float(k & 0x7FFFFFFFu)
                           : __uint_as_float(~k);
}

// Wave-local LDS handshake: all DS ops of this wave complete + no reordering.
#define WV_SYNC()                                     \
  do {                                                \
    __builtin_amdgcn_wave_barrier();                  \
    asm volatile("s_wait_dscnt 0x0" ::: "memory");    \
  } while (0)

// One wave scans a 256-bin histogram from the top and returns the bin B such
// that suffix_count(B) >= k > suffix_count(B+1).  *sAboveOut = suffix(B+1).
__device__ __forceinline__ unsigned hist_select(const unsigned* h,
                                                volatile unsigned* scr,
                                                volatile unsigned* gmax,
                                                int lane, unsigned k,
                                                unsigned* sAboveOut) {
  unsigned gs = 0;
#pragma unroll
  for (int b = 0; b < 8; ++b) gs += h[lane * 8 + b];
  scr[lane] = gs;
  WV_SYNC();
  unsigned suf = 0;
  for (int j = lane; j < 32; ++j) suf += scr[j];   // suffix sum of group sums
  scr[32 + lane] = suf;
  if (lane == 0) { scr[64] = 0u; *gmax = 0u; }
  WV_SYNC();
  atomicMax((unsigned*)gmax, (suf >= k) ? (unsigned)lane : 0u);
  WV_SYNC();
  const unsigned G = *gmax;                // highest group with suffix >= k
  unsigned sAbove = scr[32 + G + 1];       // suffix of group G+1 (0 if G==31)
  unsigned Bd = 8u * G;
  for (int b = (int)(8u * G + 7u); b >= (int)(8u * G); --b) {
    unsigned cb = h[b];
    if (sAbove + cb >= k) { Bd = (unsigned)b; break; }
    sAbove += cb;
  }
  *sAboveOut = sAbove;
  return Bd;
}

// ---------------------------------------------------------------------------
__global__ __launch_bounds__(512) void topk_phase1(const float* __restrict__ Q,
                                                   const float* __restrict__ C,
                                                   unsigned* __restrict__ wsTop) {
  __shared__ float    qs[QTILE][DIMS];            //   4 KB
  __shared__ unsigned keysLDS[QTILE * KPITCH];    //  64 KB (score keys)
  __shared__ unsigned hist[QTILE][256];           //  16 KB
  __shared__ unsigned poolLDS[QTILE][POOLCAP];    //  32 KB
  __shared__ unsigned tops[2][QTILE][KTOP];       // 12.5 KB (running top-100)
  __shared__ unsigned scr[QTILE][66];
  __shared__ unsigned gmax[QTILE];
  __shared__ unsigned cnt[QTILE];
  __shared__ unsigned poolcnt[QTILE];

  const int tid   = (int)threadIdx.x;
  const int lane  = tid & 31;
  const int w     = tid >> 5;                     // wave id 0..15
  const int qt    = (int)blockIdx.x;              // query tile 0..63
  const int chunk = (int)blockIdx.y;              // candidate chunk 0..63

  for (int i = tid; i < QTILE * DIMS; i += 512)
    qs[i >> 6][i & 63] = Q[(size_t)(qt * QTILE + (i >> 6)) * DIMS + (i & 63)];
  for (int i = tid; i < QTILE * KTOP; i += 512) (&tops[0][0][0])[i] = 0u; // -inf
  __syncthreads();

  // A fragments: Q tile, lane = (M=lane&15, Khalf=lane>>4), 16 frags cover K=64
  const int nIdx = lane & 15;
  const int kh   = (lane >> 4) << 1;              // 0 or 2
  v2f a[16];
#pragma unroll
  for (int t = 0; t < 16; ++t) {
    a[t].x = qs[nIdx][4 * t + kh];
    a[t].y = qs[nIdx][4 * t + kh + 1];
  }

  int cur = 0;
  for (int s = 0; s < NSUB; ++s) {
    __syncthreads();                               // prior selection done
    for (int i = tid; i < QTILE * 256; i += 512) (&hist[0][0])[i] = 0u;
    __syncthreads();

    // ---- compute: 64 score tiles, 4 per wave, paired + split-K chains ----
    const long candBase = (long)chunk * CHUNK + (long)s * SUBC;
    const int rbase = (lane >> 4) * 8;
    for (int i = 0; i < 4; i += 2) {
      const int t0 = w + 16 * i;
      const int t1 = w + 16 * (i + 1);
      const float* p0 = C + (size_t)(candBase + t0 * 16 + nIdx) * DIMS + kh;
      const float* p1 = C + (size_t)(candBase + t1 * 16 + nIdx) * DIMS + kh;

      // preload all 32 B-fragments -> single load clause, single wait
      v2f b0[16], b1[16];
#pragma unroll
      for (int t = 0; t < 16; ++t) {
        b0[t] = *(const v2f*)(p0 + 4 * t);
        b1[t] = *(const v2f*)(p1 + 4 * t);
      }

      // 4 independent WMMA chains (2 tiles x split-K), length 8 each
      v8f c0a = {}, c0b = {}, c1a = {}, c1b = {};
#pragma unroll
      for (int t = 0; t < 8; ++t) {
        c0a = wmma_f32_k4(a[t],     b0[t],     c0a);
        c1a = wmma_f32_k4(a[t],     b1[t],     c1a);
        c0b = wmma_f32_k4(a[t + 8], b0[t + 8], c0b);
        c1b = wmma_f32_k4(a[t + 8], b1[t + 8], c1b);
      }

      // writeback keys + fused MSB-byte histogram
      const int col0 = t0 * 16 + nIdx;
      const int col1 = t1 * 16 + nIdx;
#pragma unroll
      for (int v = 0; v < 8; ++v) {
        unsigned k0 = f2k(c0a[v] + c0b[v]);
        unsigned k1 = f2k(c1a[v] + c1b[v]);
        keysLDS[(rbase + v) * KPITCH + col0] = k0;
        keysLDS[(rbase + v) * KPITCH + col1] = k1;
        atomicAdd(&hist[rbase + v][k0 >> 24], 1u);
        atomicAdd(&hist[rbase + v][k1 >> 24], 1u);
      }
    }
    __syncthreads();                               // all scores visible

    // ---- selection: wave w does exact radix-select on row w ----
    {
      unsigned* rowKeys = &keysLDS[w * KPITCH];
      unsigned* topOld  = tops[cur][w];
      unsigned* topNew  = tops[cur ^ 1][w];
      unsigned* h       = hist[w];

      // fold running top-100 into level-0 histogram
      for (int i = lane; i < KTOP; i += 32) atomicAdd(&h[topOld[i] >> 24], 1u);
      WV_SYNC();

      unsigned k = KTOP, prefix = 0, sAbove = 0;
      const unsigned B0 = hist_select(h, scr[w], &gmax[w], lane, k, &sAbove);
      k -= sAbove;
      prefix = B0;

      // pool survivors (top byte >= B0); exact fallback on overflow
      if (lane == 0) { poolcnt[w] = 0u; cnt[w] = 0u; }
      WV_SYNC();
      for (int i = lane; i < SUBC; i += 32) {
        unsigned key = rowKeys[i];
        if ((key >> 24) >= B0) {
          unsigned p = atomicAdd(&poolcnt[w], 1u);
          if (p < POOLCAP) poolLDS[w][p] = key;
        }
      }
      for (int i = lane; i < KTOP; i += 32) {
        unsigned key = topOld[i];
        if ((key >> 24) >= B0) {
          unsigned p = atomicAdd(&poolcnt[w], 1u);
          if (p < POOLCAP) poolLDS[w][p] = key;
        }
      }
      WV_SYNC();
      const unsigned pc = poolcnt[w];
      const bool usePool = (pc <= POOLCAP);

      for (int d = 1; d < 4; ++d) {
        const int shift = 24 - 8 * d;
        for (int b = lane; b < 256; b += 32) h[b] = 0u;
        WV_SYNC();
        if (usePool) {
          for (int i = lane; i < (int)pc; i += 32) {
            unsigned key = poolLDS[w][i];
            if ((key >> (shift + 8)) == prefix)
              atomicAdd(&h[(key >> shift) & 255u], 1u);
          }
        } else {
          for (int i = lane; i < SUBC; i += 32) {
            unsigned key = rowKeys[i];
            if ((key >> (shift + 8)) == prefix)
              atomicAdd(&h[(key >> shift) & 255u], 1u);
          }
          for (int i = lane; i < KTOP; i += 32) {
            unsigned key = topOld[i];
            if ((key >> (shift + 8)) == prefix)
              atomicAdd(&h[(key >> shift) & 255u], 1u);
          }
        }
        WV_SYNC();
        const unsigned Bd = hist_select(h, scr[w], &gmax[w], lane, k, &sAbove);
        k -= sAbove;
        prefix = (prefix << 8) | Bd;
      }
      const unsigned T = prefix;   // exact 100th-largest key; count(>T)=100-k

      if (usePool) {
        for (int i = lane; i < (int)pc; i += 32) {
          unsigned key = poolLDS[w][i];
          if (key > T) topNew[atomicAdd(&cnt[w], 1u)] = key;
        }
      } else {
        for (int i = lane; i < SUBC; i += 32) {
          unsigned key = rowKeys[i];
          if (key > T) topNew[atomicAdd(&cnt[w], 1u)] = key;
        }
        for (int i = lane; i < KTOP; i += 32) {
          unsigned key = topOld[i];
          if (key > T) topNew[atomicAdd(&cnt[w], 1u)] = key;
        }
      }
      WV_SYNC();
      const unsigned c = cnt[w];   // == KTOP - k  (<= 99)
      for (int i = lane; i < KTOP; i += 32)
        if ((unsigned)i >= c) topNew[i] = T;   // exact multiset pad
    }
    cur ^= 1;
  }

  // emit this chunk's per-row top-100 keys (unsorted) to workspace
  const unsigned* fin = tops[cur][w];
  const size_t base = ((size_t)(qt * QTILE + w) * NCHUNK + (size_t)chunk) * KTOP;
  for (int i = lane; i < KTOP; i += 32) wsTop[base + i] = fin[i];
}

// ---------------------------------------------------------------------------
__global__ __launch_bounds__(32) void topk_phase2(const unsigned* __restrict__ wsTop,
                                                  float* __restrict__ out) {
  __shared__ unsigned keys2[NCHUNK * KTOP];   // 6400 keys, 25.6 KB
  __shared__ unsigned hist2[256];
  __shared__ unsigned sel[128];
  __shared__ unsigned scr2[66];
  __shared__ unsigned gmax2;
  __shared__ unsigned cnt2;

  const int lane = (int)threadIdx.x;
  const int row  = (int)blockIdx.x;
  const int N    = NCHUNK * KTOP;

  for (int b = lane; b < 256; b += 32) hist2[b] = 0u;
  WV_SYNC();
  const unsigned* src = wsTop + (size_t)row * N;
  for (int i = lane; i < N; i += 32) {
    unsigned key = src[i];
    keys2[i] = key;
    atomicAdd(&hist2[key >> 24], 1u);
  }
  WV_SYNC();

  unsigned k = KTOP, prefix = 0, sAbove = 0;
  unsigned Bd = hist_select(hist2, scr2, &gmax2, lane, k, &sAbove);
  k -= sAbove;
  prefix = Bd;
  for (int d = 1; d < 4; ++d) {
    const int shift = 24 - 8 * d;
    for (int b = lane; b < 256; b += 32) hist2[b] = 0u;
    WV_SYNC();
    for (int i = lane; i < N; i += 32) {
      unsigned key = keys2[i];
      if ((key >> (shift + 8)) == prefix)
        atomicAdd(&hist2[(key >> shift) & 255u], 1u);
    }
    WV_SYNC();
    Bd = hist_select(hist2, scr2, &gmax2, lane, k, &sAbove);
    k -= sAbove;
    prefix = (prefix << 8) | Bd;
  }
  const unsigned T = prefix;

  if (lane == 0) cnt2 = 0u;
  WV_SYNC();
  for (int i = lane; i < N; i += 32) {
    unsigned key = keys2[i];
    if (key > T) sel[atomicAdd(&cnt2, 1u)] = key;
  }
  WV_SYNC();
  const unsigned c = cnt2;                       // <= 99
  for (int i = lane; i < 128; i += 32)
    if ((unsigned)i >= c) sel[i] = (i < KTOP) ? T : 0u;
  WV_SYNC();

  // bitonic sort 128 keys, descending
  for (unsigned sz = 2; sz <= 128; sz <<= 1) {
    for (unsigned j = sz >> 1; j > 0; j >>= 1) {
      for (int i = lane; i < 128; i += 32) {
        const int p = i ^ (int)j;
        if (p > i) {
          unsigned x = sel[i], y = sel[p];
          const bool maxAtI = (((unsigned)i & sz) == 0u);
          const bool doSwap = maxAtI ? (x < y) : (x > y);
          if (doSwap) { sel[i] = y; sel[p] = x; }
        }
      }
      WV_SYNC();
    }
  }

  for (int i = lane; i < KTOP; i += 32)
    out[(size_t)row * KTOP + i] = k2f(sel[i]);
}

// ---------------------------------------------------------------------------
extern "C" void kernel_launch(void* const* d_in, const int* in_sizes, int n_in,
                              void* d_out, int out_size, void* d_ws, size_t ws_size,
                              hipStream_t stream) {
  const float* Q = (const float*)d_in[0];   // [1024, 64] fp32
  const float* C = (const float*)d_in[1];   // [1048576, 64] fp32
  float* out     = (float*)d_out;           // [1024, 100] fp32
  unsigned* wsk  = (unsigned*)d_ws;         // 1024*64*100 keys = 26.2 MB

  dim3 g1(NQT, NCHUNK);
  topk_phase1<<<g1, dim3(512), 0, stream>>>(Q, C, wsk);
  topk_phase2<<<dim3(QTILE * NQT), dim3(32), 0, stream>>>(wsk, out);
}